// TrimmedMAELoss_33406255628551
// MI455X (gfx1250) — compile-verified
//
#include <hip/hip_runtime.h>
#include <cstdint>
#include <cstddef>

// ---------------------------------------------------------------------------
// Trimmed MAE loss via exact 3-level radix select on float bit patterns.
//   pass0: digit = bits[31:21] (11b)   pass1: bits[20:10] (11b)   pass2: bits[9:0] (10b)
// Non-negative floats are order-isomorphic to their uint32 patterns, so the
// selected threshold T and tie count reproduce jnp.sort exactly.
// Fast path caches residual bit patterns (invalid -> 0xFFFFFFFF, unreachable
// by any finite |res|) in workspace so passes 1/2/sum stream 1/3 the bytes.
// ---------------------------------------------------------------------------

#define N_PIX   (512 * 512)
#define NBINS   2048
#define BLK     256
#define NSEG    32                 // blocks per image for streaming passes
#define SEG     (N_PIX / NSEG)     // 8192 elements per block
#define CHUNK   (BLK * 4)          // 1024 elements per pipelined chunk
#define NCHUNK  (SEG / CHUNK)      // 8 chunks per block
#define KEEP_F  0.8f               // == float32(1.0 - 0.2), matches jnp
#define INVALID_BITS 0xFFFFFFFFu

#if defined(__HIP_DEVICE_COMPILE__) && defined(__gfx1250__) && \
    __has_builtin(__builtin_amdgcn_global_load_async_to_lds_b128)
#define USE_ASYNC 1
#else
#define USE_ASYNC 0
#endif

#if USE_ASYNC
typedef int v4i __attribute__((ext_vector_type(4)));
typedef __attribute__((address_space(1))) v4i as1_v4i;   // global
typedef __attribute__((address_space(3))) v4i as3_v4i;   // LDS

__device__ __forceinline__ void async_cp16(const void* g, void* l) {
  // per-lane 16B global -> LDS, tracked by ASYNCcnt
  __builtin_amdgcn_global_load_async_to_lds_b128((as1_v4i*)g, (as3_v4i*)l, 0, 0);
}

template <int N>
__device__ __forceinline__ void wait_asynccnt() {
#if __has_builtin(__builtin_amdgcn_s_wait_asynccnt)
  __builtin_amdgcn_s_wait_asynccnt(N);
#else
  asm volatile("s_wait_asynccnt %0" ::"n"(N) : "memory");
#endif
  asm volatile("" ::: "memory");  // keep LDS reads below the wait
}
#endif

// ---------------------------------------------------------------------------
__global__ void zero_u32_kernel(unsigned* __restrict__ p, int n) {
  int i = blockIdx.x * blockDim.x + threadIdx.x;
  if (i < n) p[i] = 0u;
}

// ===========================================================================
// FAST PATH kernels (residual bits cached in workspace)
// ===========================================================================

// pass 0 + residual-cache write: stream pred/targ/mask, emit packed bits,
// histogram top 11 bits of valid residuals.
__global__ void pass0_cache_kernel(const float* __restrict__ pred,
                                   const float* __restrict__ targ,
                                   const int* __restrict__ mask,
                                   unsigned* __restrict__ resBits,
                                   unsigned* __restrict__ hist) {
  __shared__ unsigned lh[NBINS];
  const int tid = threadIdx.x;
  const int b = blockIdx.y;
  for (int i = tid; i < NBINS; i += BLK) lh[i] = 0u;
  __syncthreads();

  const size_t base = (size_t)b * (size_t)N_PIX + (size_t)blockIdx.x * SEG;

#if USE_ASYNC
  __shared__ float bufP[2][CHUNK];
  __shared__ float bufT[2][CHUNK];
  __shared__ int   bufM[2][CHUNK];
  {
    size_t g = base + (size_t)tid * 4;
    async_cp16(pred + g, &bufP[0][tid * 4]);
    async_cp16(targ + g, &bufT[0][tid * 4]);
    async_cp16(mask + g, &bufM[0][tid * 4]);
  }
  for (int c = 0; c < NCHUNK; ++c) {
    const int cur = c & 1;
    if (c + 1 < NCHUNK) {
      const int nxt = cur ^ 1;
      size_t g = base + (size_t)(c + 1) * CHUNK + (size_t)tid * 4;
      async_cp16(pred + g, &bufP[nxt][tid * 4]);
      async_cp16(targ + g, &bufT[nxt][tid * 4]);
      async_cp16(mask + g, &bufM[nxt][tid * 4]);
      wait_asynccnt<3>();
    } else {
      wait_asynccnt<0>();
    }
    uint4 ov;
    unsigned* op = (unsigned*)&ov;
#pragma unroll
    for (int j = 0; j < 4; ++j) {
      const int li = tid * 4 + j;
      unsigned v = INVALID_BITS;
      if (bufM[cur][li] > 0) {
        v = __float_as_uint(fabsf(bufP[cur][li] - bufT[cur][li]));
        atomicAdd(&lh[v >> 21], 1u);
      }
      op[j] = v;
    }
    *(uint4*)(resBits + base + (size_t)c * CHUNK + (size_t)tid * 4) = ov;
  }
#else
  for (int c = 0; c < NCHUNK; ++c) {
    size_t g = base + (size_t)c * CHUNK + (size_t)tid * 4;
    const float4 p4 = *(const float4*)(pred + g);
    const float4 t4 = *(const float4*)(targ + g);
    const int4   m4 = *(const int4*)(mask + g);
    const float pv[4] = {p4.x, p4.y, p4.z, p4.w};
    const float tv[4] = {t4.x, t4.y, t4.z, t4.w};
    const int   mv[4] = {m4.x, m4.y, m4.z, m4.w};
    uint4 ov;
    unsigned* op = (unsigned*)&ov;
#pragma unroll
    for (int j = 0; j < 4; ++j) {
      unsigned v = INVALID_BITS;
      if (mv[j] > 0) {
        v = __float_as_uint(fabsf(pv[j] - tv[j]));
        atomicAdd(&lh[v >> 21], 1u);
      }
      op[j] = v;
    }
    *(uint4*)(resBits + g) = ov;
  }
#endif

  __syncthreads();
  unsigned* gh = hist + (size_t)b * NBINS;
  for (int i = tid; i < NBINS; i += BLK) {
    unsigned cnt = lh[i];
    if (cnt) atomicAdd(&gh[i], cnt);
  }
}

// refine pass over cached bits (single stream): histogram one digit among
// elements whose high bits match the image's selected prefix.
__global__ void refine_cache_kernel(const unsigned* __restrict__ resBits,
                                    unsigned* __restrict__ hist,
                                    const unsigned* __restrict__ selPrefix,
                                    int digit_shift, unsigned digit_mask,
                                    int prefix_shift) {
  __shared__ unsigned lh[NBINS];
  const int tid = threadIdx.x;
  const int b = blockIdx.y;
  const unsigned sel = selPrefix[b];
  for (int i = tid; i < NBINS; i += BLK) lh[i] = 0u;
  __syncthreads();

  const size_t base = (size_t)b * (size_t)N_PIX + (size_t)blockIdx.x * SEG;

#if USE_ASYNC
  __shared__ unsigned bufV[2][CHUNK];
  async_cp16(resBits + base + (size_t)tid * 4, &bufV[0][tid * 4]);
  for (int c = 0; c < NCHUNK; ++c) {
    const int cur = c & 1;
    if (c + 1 < NCHUNK) {
      async_cp16(resBits + base + (size_t)(c + 1) * CHUNK + (size_t)tid * 4,
                 &bufV[cur ^ 1][tid * 4]);
      wait_asynccnt<1>();
    } else {
      wait_asynccnt<0>();
    }
#pragma unroll
    for (int j = 0; j < 4; ++j) {
      unsigned v = bufV[cur][tid * 4 + j];
      if ((v >> prefix_shift) == sel)
        atomicAdd(&lh[(v >> digit_shift) & digit_mask], 1u);
    }
  }
#else
  for (int c = 0; c < NCHUNK; ++c) {
    const uint4 v4 = *(const uint4*)(resBits + base + (size_t)c * CHUNK +
                                     (size_t)tid * 4);
    const unsigned vv[4] = {v4.x, v4.y, v4.z, v4.w};
#pragma unroll
    for (int j = 0; j < 4; ++j) {
      unsigned v = vv[j];
      if ((v >> prefix_shift) == sel)
        atomicAdd(&lh[(v >> digit_shift) & digit_mask], 1u);
    }
  }
#endif

  __syncthreads();
  unsigned* gh = hist + (size_t)b * NBINS;
  for (int i = tid; i < NBINS; i += BLK) {
    unsigned cnt = lh[i];
    if (cnt) atomicAdd(&gh[i], cnt);
  }
}

// sum of cached residuals strictly below threshold (unsigned bit compare ==
// float compare; INVALID_BITS can never be < any threshold of a valid value).
__global__ void sum_cache_kernel(const unsigned* __restrict__ resBits,
                                 const unsigned* __restrict__ Tbits,
                                 float* __restrict__ partial) {
  const int tid = threadIdx.x;
  const int b = blockIdx.y;
  const unsigned T = Tbits[b];
  const size_t base = (size_t)b * (size_t)N_PIX + (size_t)blockIdx.x * SEG;
  float acc = 0.0f;

#if USE_ASYNC
  __shared__ unsigned bufV[2][CHUNK];
  async_cp16(resBits + base + (size_t)tid * 4, &bufV[0][tid * 4]);
  for (int c = 0; c < NCHUNK; ++c) {
    const int cur = c & 1;
    if (c + 1 < NCHUNK) {
      async_cp16(resBits + base + (size_t)(c + 1) * CHUNK + (size_t)tid * 4,
                 &bufV[cur ^ 1][tid * 4]);
      wait_asynccnt<1>();
    } else {
      wait_asynccnt<0>();
    }
#pragma unroll
    for (int j = 0; j < 4; ++j) {
      unsigned v = bufV[cur][tid * 4 + j];
      if (v < T) acc += __uint_as_float(v);
    }
  }
#else
  for (int c = 0; c < NCHUNK; ++c) {
    const uint4 v4 = *(const uint4*)(resBits + base + (size_t)c * CHUNK +
                                     (size_t)tid * 4);
    const unsigned vv[4] = {v4.x, v4.y, v4.z, v4.w};
#pragma unroll
    for (int j = 0; j < 4; ++j)
      if (vv[j] < T) acc += __uint_as_float(vv[j]);
  }
#endif

  for (int off = 16; off > 0; off >>= 1) acc += __shfl_down(acc, off, 32);
  __shared__ float red[BLK / 32];
  const int lane = tid & 31, wid = tid >> 5;
  if (lane == 0) red[wid] = acc;
  __syncthreads();
  if (tid == 0) {
    float s = 0.0f;
    for (int w = 0; w < BLK / 32; ++w) s += red[w];
    partial[(size_t)b * gridDim.x + blockIdx.x] = s;
  }
}

// ===========================================================================
// FALLBACK kernels (recompute residual from inputs; used if ws too small)
// ===========================================================================

__global__ void hist_pass_kernel(const float* __restrict__ pred,
                                 const float* __restrict__ targ,
                                 const int* __restrict__ mask,
                                 unsigned* __restrict__ hist,
                                 const unsigned* __restrict__ selPrefix,
                                 int digit_shift, unsigned digit_mask,
                                 int prefix_shift) {
  __shared__ unsigned lh[NBINS];
  const int tid = threadIdx.x;
  const int b = blockIdx.y;
  const unsigned sel = (prefix_shift < 32) ? selPrefix[b] : 0u;
  for (int i = tid; i < NBINS; i += BLK) lh[i] = 0u;
  __syncthreads();

  const size_t base = (size_t)b * (size_t)N_PIX + (size_t)blockIdx.x * SEG;
  for (int c = 0; c < NCHUNK; ++c) {
    size_t g = base + (size_t)c * CHUNK + (size_t)tid * 4;
    const float4 p4 = *(const float4*)(pred + g);
    const float4 t4 = *(const float4*)(targ + g);
    const int4   m4 = *(const int4*)(mask + g);
    const float pv[4] = {p4.x, p4.y, p4.z, p4.w};
    const float tv[4] = {t4.x, t4.y, t4.z, t4.w};
    const int   mv[4] = {m4.x, m4.y, m4.z, m4.w};
#pragma unroll
    for (int j = 0; j < 4; ++j) {
      if (mv[j] > 0) {
        unsigned v = __float_as_uint(fabsf(pv[j] - tv[j]));
        if (prefix_shift >= 32 || (v >> prefix_shift) == sel)
          atomicAdd(&lh[(v >> digit_shift) & digit_mask], 1u);
      }
    }
  }

  __syncthreads();
  unsigned* gh = hist + (size_t)b * NBINS;
  for (int i = tid; i < NBINS; i += BLK) {
    unsigned cnt = lh[i];
    if (cnt) atomicAdd(&gh[i], cnt);
  }
}

__global__ void sum_pass_kernel(const float* __restrict__ pred,
                                const float* __restrict__ targ,
                                const int* __restrict__ mask,
                                const unsigned* __restrict__ Tbits,
                                float* __restrict__ partial) {
  const int tid = threadIdx.x;
  const int b = blockIdx.y;
  const unsigned T = Tbits[b];
  const size_t base = (size_t)b * (size_t)N_PIX + (size_t)blockIdx.x * SEG;
  float acc = 0.0f;

  for (int c = 0; c < NCHUNK; ++c) {
    size_t g = base + (size_t)c * CHUNK + (size_t)tid * 4;
    const float4 p4 = *(const float4*)(pred + g);
    const float4 t4 = *(const float4*)(targ + g);
    const int4   m4 = *(const int4*)(mask + g);
    const float pv[4] = {p4.x, p4.y, p4.z, p4.w};
    const float tv[4] = {t4.x, t4.y, t4.z, t4.w};
    const int   mv[4] = {m4.x, m4.y, m4.z, m4.w};
#pragma unroll
    for (int j = 0; j < 4; ++j) {
      if (mv[j] > 0) {
        float r = fabsf(pv[j] - tv[j]);
        if (__float_as_uint(r) < T) acc += r;
      }
    }
  }

  for (int off = 16; off > 0; off >>= 1) acc += __shfl_down(acc, off, 32);
  __shared__ float red[BLK / 32];
  const int lane = tid & 31, wid = tid >> 5;
  if (lane == 0) red[wid] = acc;
  __syncthreads();
  if (tid == 0) {
    float s = 0.0f;
    for (int w = 0; w < BLK / 32; ++w) s += red[w];
    partial[(size_t)b * gridDim.x + blockIdx.x] = s;
  }
}

// ===========================================================================
// Shared small kernels
// ===========================================================================

__global__ void scan_kernel(const unsigned* __restrict__ hist,
                            unsigned* __restrict__ selPrefix,
                            unsigned* __restrict__ remaining,
                            unsigned* __restrict__ Mcnt,
                            unsigned* __restrict__ Tbits,
                            unsigned* __restrict__ takeCnt,
                            int pass, int nbits, int B) {
  const int b = blockIdx.x * blockDim.x + threadIdx.x;
  if (b >= B) return;
  const unsigned* h = hist + (size_t)b * NBINS;
  const int nb = 1 << nbits;

  if (pass == 0) {
    unsigned total = 0;
    for (int i = 0; i < nb; ++i) total += h[i];
    Mcnt[b] = total;
    const unsigned k = (unsigned)floorf((float)total * KEEP_F);
    if (k == 0) { selPrefix[b] = INVALID_BITS; remaining[b] = 0u; return; }
    unsigned cum = 0, sel = 0, rem = 0; bool found = false;
    for (int i = 0; i < nb; ++i) {
      unsigned c = h[i];
      if (!found && cum + c >= k) { sel = (unsigned)i; rem = k - cum; found = true; }
      cum += c;
    }
    selPrefix[b] = sel;
    remaining[b] = rem;
  } else {
    const unsigned sp = selPrefix[b];
    if (sp == INVALID_BITS) {  // empty image: keep nothing
      if (pass == 2) { Tbits[b] = 0u; takeCnt[b] = 0u; }
      return;
    }
    const unsigned k = remaining[b];
    unsigned cum = 0, sel = 0, rem = 0; bool found = false;
    for (int i = 0; i < nb; ++i) {
      unsigned c = h[i];
      if (!found && cum + c >= k) { sel = (unsigned)i; rem = k - cum; found = true; }
      cum += c;
    }
    const unsigned np = (sp << nbits) | sel;
    selPrefix[b] = np;
    remaining[b] = rem;
    if (pass == 2) { Tbits[b] = np; takeCnt[b] = rem; }
  }
}

__global__ void finalize_kernel(const float* __restrict__ partial,
                                const unsigned* __restrict__ Tbits,
                                const unsigned* __restrict__ takeCnt,
                                const unsigned* __restrict__ Mcnt,
                                float* __restrict__ out, int B, int nseg) {
  __shared__ float sh_loss[32];
  __shared__ float sh_div[32];
  const int t = threadIdx.x;
  float loss = 0.0f, divi = 0.0f;
  for (int b = t; b < B; b += 32) {
    float s = 0.0f;
    for (int j = 0; j < nseg; ++j) s += partial[(size_t)b * nseg + j];
    s += (float)takeCnt[b] * __uint_as_float(Tbits[b]);  // tie copies at T
    loss += s;
    divi += (float)Mcnt[b] * KEEP_F;
  }
  sh_loss[t] = loss;
  sh_div[t] = divi;
  __syncthreads();
  if (t == 0) {
    float tot = 0.0f, dv = 0.0f;
    for (int i = 0; i < 32; ++i) { tot += sh_loss[i]; dv += sh_div[i]; }
    out[0] = (dv == 0.0f) ? 0.0f : tot / fmaxf(dv, 1e-30f);
  }
}

// ---------------------------------------------------------------------------
extern "C" void kernel_launch(void* const* d_in, const int* in_sizes, int n_in,
                              void* d_out, int out_size, void* d_ws, size_t ws_size,
                              hipStream_t stream) {
  const float* pred = (const float*)d_in[0];
  const float* targ = (const float*)d_in[1];
  const int*   mask = (const int*)d_in[2];
  float* out = (float*)d_out;
  const int B = in_sizes[0] / N_PIX;

  const size_t cacheWords = (size_t)B * N_PIX;                 // residual bits
  const size_t stateWords = (size_t)B * NBINS + 5 * (size_t)B  // hist + state
                          + (size_t)B * NSEG;                  // partials
  const bool fast = ws_size >= (cacheWords + stateWords) * sizeof(unsigned);

  unsigned* resBits   = (unsigned*)d_ws;                       // fast path only
  unsigned* hist      = fast ? resBits + cacheWords : (unsigned*)d_ws;
  unsigned* selPrefix = hist + (size_t)B * NBINS;
  unsigned* remaining = selPrefix + B;
  unsigned* Mcnt      = remaining + B;
  unsigned* Tbits     = Mcnt + B;
  unsigned* takeCnt   = Tbits + B;
  float*    partial   = (float*)(takeCnt + B);

  const dim3 gridH(NSEG, B);
  const int histN = B * NBINS;
  const dim3 gz((histN + BLK - 1) / BLK);
  const dim3 gscan((B + 31) / 32);

  if (fast) {
    // pass 0: histogram bits[31:21] + write packed residual bits
    zero_u32_kernel<<<gz, BLK, 0, stream>>>(hist, histN);
    pass0_cache_kernel<<<gridH, BLK, 0, stream>>>(pred, targ, mask, resBits, hist);
    scan_kernel<<<gscan, 32, 0, stream>>>(hist, selPrefix, remaining, Mcnt,
                                          Tbits, takeCnt, 0, 11, B);
    // pass 1: digit bits[20:10], prefix bits[31:21]
    zero_u32_kernel<<<gz, BLK, 0, stream>>>(hist, histN);
    refine_cache_kernel<<<gridH, BLK, 0, stream>>>(resBits, hist, selPrefix,
                                                   10, 0x7FFu, 21);
    scan_kernel<<<gscan, 32, 0, stream>>>(hist, selPrefix, remaining, Mcnt,
                                          Tbits, takeCnt, 1, 11, B);
    // pass 2: digit bits[9:0], prefix bits[31:10]
    zero_u32_kernel<<<gz, BLK, 0, stream>>>(hist, histN);
    refine_cache_kernel<<<gridH, BLK, 0, stream>>>(resBits, hist, selPrefix,
                                                   0, 0x3FFu, 10);
    scan_kernel<<<gscan, 32, 0, stream>>>(hist, selPrefix, remaining, Mcnt,
                                          Tbits, takeCnt, 2, 10, B);
    // sum below T, finalize with tie copies + divisor
    sum_cache_kernel<<<gridH, BLK, 0, stream>>>(resBits, Tbits, partial);
  } else {
    zero_u32_kernel<<<gz, BLK, 0, stream>>>(hist, histN);
    hist_pass_kernel<<<gridH, BLK, 0, stream>>>(pred, targ, mask, hist,
                                                selPrefix, 21, 0x7FFu, 32);
    scan_kernel<<<gscan, 32, 0, stream>>>(hist, selPrefix, remaining, Mcnt,
                                          Tbits, takeCnt, 0, 11, B);
    zero_u32_kernel<<<gz, BLK, 0, stream>>>(hist, histN);
    hist_pass_kernel<<<gridH, BLK, 0, stream>>>(pred, targ, mask, hist,
                                                selPrefix, 10, 0x7FFu, 21);
    scan_kernel<<<gscan, 32, 0, stream>>>(hist, selPrefix, remaining, Mcnt,
                                          Tbits, takeCnt, 1, 11, B);
    zero_u32_kernel<<<gz, BLK, 0, stream>>>(hist, histN);
    hist_pass_kernel<<<gridH, BLK, 0, stream>>>(pred, targ, mask, hist,
                                                selPrefix, 0, 0x3FFu, 10);
    scan_kernel<<<gscan, 32, 0, stream>>>(hist, selPrefix, remaining, Mcnt,
                                          Tbits, takeCnt, 2, 10, B);
    sum_pass_kernel<<<gridH, BLK, 0, stream>>>(pred, targ, mask, Tbits, partial);
  }
  finalize_kernel<<<1, 32, 0, stream>>>(partial, Tbits, takeCnt, Mcnt, out, B, NSEG);
}